// BoundedReLU_59949153518085
// MI455X (gfx1250) — compile-verified
//
#include <hip/hip_runtime.h>

// BoundedReLU relaxation: elementwise bounds + dense-diagonal coefficient
// matrices. 538 MB written / ~0.5 MB read -> pure store-bandwidth problem.
// Split into (1) a minimal-overhead 512 MB NT zero-stream and (2) a tiny
// kernel that computes the relaxation and patches diagonals + small outputs.

typedef __attribute__((ext_vector_type(4))) float v4f;

#define BB 64
#define NN 1024
#define EPSF 1e-8f

// ---- Kernel 1: zero-fill the contiguous [2 x B x N x N] coef region ----
// 2048 blocks x 256 threads; each block owns a contiguous 256 KB span.
// Each thread: 64 unrolled NT b128 stores at 4 KB stride (imm offsets off one
// base). Consecutive lanes -> consecutive float4s: 512 B/wave per store instr,
// 32 KB stored per wave for ~80 instructions of total overhead.
#define FILL_BLOCKS  2048
#define FILL_THREADS 256
#define FILL_ITERS   64   // (2*64*1024*1024 / 4 float4) / (2048*256 threads)

__global__ __launch_bounds__(FILL_THREADS) void coef_zero_fill(
    float* __restrict__ coef)   // points at out + 2*B*N (start of lower_coef)
{
    const size_t blockBase =
        (size_t)blockIdx.x * ((size_t)FILL_THREADS * FILL_ITERS * 4);
    float* p = coef + blockBase + (size_t)threadIdx.x * 4;

    const v4f z = (v4f){0.0f, 0.0f, 0.0f, 0.0f};
#pragma unroll
    for (int it = 0; it < FILL_ITERS; ++it) {
        // offset step = 256 threads * 16 B = 4096 B -> stays in imm offset
        __builtin_nontemporal_store(z, (v4f*)(p + (size_t)it * (FILL_THREADS * 4)));
    }
}

// ---- Kernel 2: relaxation scalars, small [B,N] outputs, diagonal patch ----
__global__ __launch_bounds__(256) void relax_diag(
    const float* __restrict__ lower,
    const float* __restrict__ upper,
    float* __restrict__ out)
{
    const unsigned row = blockIdx.x * 256 + threadIdx.x;  // 0 .. B*N-1
    const unsigned i   = row & (NN - 1);                  // neuron index

    const float l = lower[row];
    const float u = upper[row];

    const bool  active   = (l >= 0.0f);
    const bool  crossing = (l < 0.0f) && (u > 0.0f);
    const float lam      = crossing ? (u / (u - l + EPSF)) : 0.0f;
    const float mu       = crossing ? (-lam * l)           : 0.0f;
    const float ldiag    = active ? 1.0f : 0.0f;
    const float udiag    = active ? 1.0f : lam;           // lam==0 when inactive

    // Output layout = reference tuple order, concatenated flat.
    const size_t BN  = (size_t)BB * NN;                   // 65536
    const size_t BNN = BN * NN;                           // 67108864
    float* conc_l   = out;                                // [B,N]
    float* conc_u   = out + BN;                           // [B,N]
    float* low_coef = out + 2 * BN;                       // [B,N,N]
    float* up_coef  = out + 2 * BN + BNN;                 // [B,N,N]
    float* low_bias = out + 2 * BN + 2 * BNN;             // [B,N]
    float* up_bias  = low_bias + BN;                      // [B,N]

    // Coalesced small outputs.
    conc_l[row]   = fmaxf(l, 0.0f);
    conc_u[row]   = fmaxf(u, 0.0f);
    low_bias[row] = 0.0f;
    up_bias[row]  = mu;

    // Diagonal element (b, i, i): flat offset = row*N + i. Scattered 4 B
    // stores, one line each; runs after the zero fill via stream ordering.
    const size_t d = (size_t)row * NN + i;
    low_coef[d] = ldiag;
    up_coef[d]  = udiag;
}

extern "C" void kernel_launch(void* const* d_in, const int* in_sizes, int n_in,
                              void* d_out, int out_size, void* d_ws, size_t ws_size,
                              hipStream_t stream)
{
    const float* lower = (const float*)d_in[0];
    const float* upper = (const float*)d_in[1];
    float* out = (float*)d_out;

    const size_t BN = (size_t)BB * NN;
    float* coef = out + 2 * BN;   // contiguous lower_coef ++ upper_coef region

    // 1) Stream 512 MB of zeros at HBM rate.
    coef_zero_fill<<<dim3(FILL_BLOCKS), dim3(FILL_THREADS), 0, stream>>>(coef);
    // 2) Patch diagonals + write the four small [B,N] outputs (WAW ordered on
    //    the stream, preserved by graph capture).
    relax_diag<<<dim3((BB * NN) / 256), dim3(256), 0, stream>>>(lower, upper, out);
}